// ConvCapsuleLayer_53034256171604
// MI455X (gfx1250) — compile-verified
//
#include <hip/hip_runtime.h>

typedef __attribute__((ext_vector_type(2))) float v2f;
typedef __attribute__((ext_vector_type(8))) float v8f;

#define IN_B 16
#define IN_H 32
#define IN_W 32
#define IN_C 8
#define IN_A 16
#define OUTC 8
#define OUTA 32

// LDS layout (float indices into dynamic smem):
//   votesS : [0,     32768)   8 img * 16 pos * 256 oc        (128 KB)
//   inT    : [32768, 45568)   8 img * 5 row * 20 col * 16 ch (50 KB)
//   wbuf   : [45568, 53760)   2 buffers * (16 k x 256 n)     (32 KB)
//            k-pair interleaved: (k,n) -> ((k>>1)*256 + n)*2 + (k&1)
//   logitsS: [53760, 54784)   16 pos * 8 i * 8 c             (4 KB)
//   routeS : [54784, 55808)                                   (4 KB)
// total 55808 floats = 223232 bytes (< 320 KB WGP LDS)
#define VOTES_OFF  0
#define INT_OFF    32768
#define WBUF_OFF   45568
#define LOGITS_OFF 53760
#define ROUTE_OFF  54784

__device__ __forceinline__ void async_wait0() {
  asm volatile("s_wait_asynccnt 0x0" ::: "memory");
}

// Stage one 16x256 fp32 W block (16 KB) into LDS buffer `buf` asynchronously,
// k-pair interleaving on the fly (async-to-LDS has per-lane addresses on both
// global and LDS sides, so the transpose is free).  Op j: the 256 threads read
// row k=j fully coalesced (consecutive 4B per lane) and scatter to the
// interleaved layout.  Tracked by ASYNCcnt.
__device__ __forceinline__ void stage_wblock_async(const float* __restrict__ Wg,
                                                   int pix, int buf, int tid) {
  const float* g = Wg + pix * 4096 + tid;
  const unsigned base =
      (unsigned)(WBUF_OFF + buf * 4096 + tid * 2) * 4u;  // bytes
#pragma unroll
  for (int j = 0; j < 16; ++j) {
    const unsigned lds = base + (unsigned)((j >> 1) * 512 + (j & 1)) * 4u;
    asm volatile("global_load_async_to_lds_b32 %0, %1, off"
                 :: "v"(lds), "v"(g + j * 256)
                 : "memory");
  }
}

extern "C" __global__ __launch_bounds__(256)
void capsconv_route(const float* __restrict__ inp,
                    const float* __restrict__ Wg,
                    const float* __restrict__ bias,
                    float* __restrict__ out) {
  extern __shared__ float smem[];
  float* votesS  = smem + VOTES_OFF;
  float* inT     = smem + INT_OFF;
  float* logitsS = smem + LOGITS_OFF;
  float* routeS  = smem + ROUTE_OFF;

  const int tid   = threadIdx.x;
  const int wave  = tid >> 5;   // i2: 0..7 (votes input-capsule slot)
  const int lane  = tid & 31;
  const int mrow  = lane & 15;  // M row (pixel in tile) / N col
  const int khalf = lane >> 4;  // K-pair select per WMMA f32 16x16x4 layout

  const int b2    = blockIdx.x >> 6;   // 0..15
  const int tile  = blockIdx.x & 63;   // 64 tiles of 16 pixels
  const int hrow  = tile >> 1;         // 0..31
  const int wbase = (tile & 1) << 4;   // 0 or 16

  // zero routing logits
  for (int t = tid; t < 1024; t += 256) logitsS[t] = 0.0f;

  // ---- stage 5x20x16 input halo tile for this wave's image ----
  // votes[b2][i2] <-> folded image n = b2*8 + i2 taken from
  // inputs[batch = n%16, :, :, cap = n/16, :]
  {
    const int nc   = b2 * 8 + wave;
    const int bimg = nc & 15;
    const int icap = nc >> 4;
    for (int idx = lane; idx < 5 * 20 * 16; idx += 32) {
      const int ia  = idx & 15;
      const int t16 = idx >> 4;
      const int col = t16 % 20;
      const int kh  = t16 / 20;
      const int hin = hrow + kh - 2;
      const int win = wbase + col - 2;
      float v = 0.0f;
      if (hin >= 0 && hin < IN_H && win >= 0 && win < IN_W)
        v = inp[(((bimg * IN_H + hin) * IN_W + win) * IN_C + icap) * IN_A + ia];
      inT[((wave * 5 + kh) * 20 + col) * 16 + ia] = v;
    }
  }

  // prime first W block (pixel 0) asynchronously
  stage_wblock_async(Wg, 0, 0, tid);
  async_wait0();
  __syncthreads();

  // ---- conv as GEMM: per wave M=16 (pixels) x N=256 (oc) x K=400 ----
  // All 16 N-tile accumulators live (128 VGPRs/lane).  For each of the 25
  // filter taps: B block (16 K x 256 N) comes from the LDS double buffer
  // (async-prefetched, k-pair interleaved so each lane's B operand is one
  // aligned b64), A comes from the LDS input tile, hoisted into regs.
  // B operands are software-pipelined in registers (2x16 v2f) so a full
  // K-step of ds_loads stays in flight behind a full K-step of WMMAs.
  v8f acc[16];
#pragma unroll
  for (int nt = 0; nt < 16; ++nt)
    acc[nt] = (v8f){0.f, 0.f, 0.f, 0.f, 0.f, 0.f, 0.f, 0.f};

#pragma unroll 1
  for (int pix = 0; pix < 25; ++pix) {
    const int cur = pix & 1;
    if (pix < 24) stage_wblock_async(Wg, pix + 1, cur ^ 1, tid);

    const int kh = pix / 5;
    const int kw = pix - kh * 5;
    const float* arow = &inT[((wave * 5 + kh) * 20 + (mrow + kw)) * 16];
    v2f a[4];
#pragma unroll
    for (int s = 0; s < 4; ++s)                 // ds_load_b64, 8B aligned
      a[s] = *(const v2f*)(arow + s * 4 + khalf * 2);

    // lane's B pair (k0 = s*4 + 2*khalf, k0+1) at interleaved float index
    // 2*((s*2 + khalf)*256 + nt*16 + mrow)
    const float* bbase =
        smem + WBUF_OFF + cur * 4096 + (khalf * 256 + mrow) * 2;

    v2f b[2][16];
#pragma unroll
    for (int nt = 0; nt < 16; ++nt)
      b[0][nt] = *(const v2f*)(bbase + (nt * 16) * 2);

#pragma unroll
    for (int s = 0; s < 4; ++s) {
      if (s < 3) {
#pragma unroll
        for (int nt = 0; nt < 16; ++nt)
          b[(s + 1) & 1][nt] =
              *(const v2f*)(bbase + ((s + 1) * 512 + nt * 16) * 2);
      }
#pragma unroll
      for (int nt = 0; nt < 16; ++nt) {
        acc[nt] = __builtin_amdgcn_wmma_f32_16x16x4_f32(
            false, a[s], false, b[s & 1][nt], (short)0, acc[nt], false, false);
      }
    }

    if (pix < 24) {
      async_wait0();    // next block fully landed in LDS (this wave's ops)
      __syncthreads();  // all waves done reading `cur`, all staging visible
    }
  }

  // C/D layout: vgpr r -> pos = r + 8*(lane/16), N = lane%16
#pragma unroll
  for (int nt = 0; nt < 16; ++nt) {
    const int n = nt * 16 + mrow;
#pragma unroll
    for (int r = 0; r < 8; ++r) {
      const int pos = r + 8 * khalf;
      votesS[(wave * 16 + pos) * 256 + n] = acc[nt][r];
    }
  }
  __syncthreads();

  // ---- dynamic routing (3 iterations) entirely out of LDS ----
  const int pos  = tid >> 4;   // pixel in tile
  const int sub  = tid & 15;
  const int c    = sub >> 1;   // out capsule
  const int half = sub & 1;    // atom half (pairs are adjacent lanes)
  const int a0   = half << 4;

  float bl[16];
#pragma unroll
  for (int j = 0; j < 16; ++j) bl[j] = bias[c * 32 + a0 + j];

  float act[16];
  for (int iter = 0; iter < 3; ++iter) {
    // softmax over out-capsules: thread sub<8 handles input capsule i=sub
    if (sub < 8) {
      const int i = sub;
      float l[8];
      float mx = -3.4e38f;
#pragma unroll
      for (int cc = 0; cc < 8; ++cc) {
        l[cc] = logitsS[(pos * 8 + i) * 8 + cc];
        mx = fmaxf(mx, l[cc]);
      }
      float s = 0.f;
#pragma unroll
      for (int cc = 0; cc < 8; ++cc) { l[cc] = expf(l[cc] - mx); s += l[cc]; }
      const float inv = 1.f / s;
#pragma unroll
      for (int cc = 0; cc < 8; ++cc) routeS[(pos * 8 + i) * 8 + cc] = l[cc] * inv;
    }
    __syncthreads();

    // preact = sum_i route[i][c] * votes[i][pos][c][a] + bias
    float pre[16];
#pragma unroll
    for (int j = 0; j < 16; ++j) pre[j] = bl[j];
    for (int i = 0; i < 8; ++i) {
      const float r = routeS[(pos * 8 + i) * 8 + c];
      const float4* vp =
          (const float4*)&votesS[(i * 16 + pos) * 256 + c * 32 + a0];
#pragma unroll
      for (int q = 0; q < 4; ++q) {
        float4 v = vp[q];                        // ds_load_b128
        pre[q * 4 + 0] += r * v.x;
        pre[q * 4 + 1] += r * v.y;
        pre[q * 4 + 2] += r * v.z;
        pre[q * 4 + 3] += r * v.w;
      }
    }
    // squash: ||v||^2 combined across the two atom-halves (lane pair)
    float nsq = 0.f;
#pragma unroll
    for (int j = 0; j < 16; ++j) nsq += pre[j] * pre[j];
    nsq += __shfl_xor(nsq, 1, 32);
    const float scale = nsq / ((1.f + nsq) * sqrtf(nsq));
#pragma unroll
    for (int j = 0; j < 16; ++j) act[j] = pre[j] * scale;

    if (iter < 2) {
      // agreement: logits[i][c] += dot(votes[i][pos][c][:], act[c][:])
      for (int i = 0; i < 8; ++i) {
        const float4* vp =
            (const float4*)&votesS[(i * 16 + pos) * 256 + c * 32 + a0];
        float p = 0.f;
#pragma unroll
        for (int q = 0; q < 4; ++q) {
          float4 v = vp[q];
          p += v.x * act[q * 4 + 0] + v.y * act[q * 4 + 1] +
               v.z * act[q * 4 + 2] + v.w * act[q * 4 + 3];
        }
        p += __shfl_xor(p, 1, 32);
        if (half == 0) logitsS[(pos * 8 + i) * 8 + c] += p;
      }
      __syncthreads();
    }
  }

  // final activations -> out[b2][h][w][c][a]
  float* op = out + (((b2 * 32 + hrow) * 32 + (wbase + pos)) * OUTC + c) * OUTA + a0;
#pragma unroll
  for (int j = 0; j < 16; ++j) op[j] = act[j];
}

extern "C" void kernel_launch(void* const* d_in, const int* in_sizes, int n_in,
                              void* d_out, int out_size, void* d_ws, size_t ws_size,
                              hipStream_t stream) {
  const float* inp  = (const float*)d_in[0];  // [16,32,32,8,16]
  const float* Wg   = (const float*)d_in[1];  // [5,5,16,256]
  const float* bias = (const float*)d_in[2];  // [1,1,8,32]
  float* out = (float*)d_out;                 // [16,32,32,8,32]

  const size_t shmem = (size_t)55808 * sizeof(float);
  // 16 batch groups * 64 spatial tiles = 1024 workgroups, 256 threads (8 waves)
  capsconv_route<<<dim3(1024), dim3(256), shmem, stream>>>(inp, Wg, bias, out);
}